// GAT_44495861186578
// MI455X (gfx1250) — compile-verified
//
#include <hip/hip_runtime.h>
#include <hip/hip_bf16.h>
#include <math.h>

#define DIN   128
#define DOUT  128
#define NHEAD 4
#define DS    32

typedef __attribute__((ext_vector_type(2))) float v2f;
typedef __attribute__((ext_vector_type(8))) float v8f;

__device__ __forceinline__ float wave_sum(float v) {
#pragma unroll
  for (int m = 16; m > 0; m >>= 1) v += __shfl_xor(v, m, 32);
  return v;
}
__device__ __forceinline__ float wave_max(float v) {
#pragma unroll
  for (int m = 16; m > 0; m >>= 1) v = fmaxf(v, __shfl_xor(v, m, 32));
  return v;
}
__device__ __forceinline__ float lrelu(float x) { return x > 0.f ? x : 0.2f * x; }

// ---------------------------------------------------------------------------
// Kernel 1: h_self = relu(A @ W0^T + b0), h_neigh = relu(A @ W1^T + b1)
// One wave per 16-row strip, covering all 128 output columns of BOTH GEMMs,
// so the A fragment is loaded once per K-step and reused for 16 WMMAs.
// fp32 WMMA (16x16x4): memory-bound problem, keep full reference precision.
// EXEC must be all-ones for WMMA: tail rows are handled by clamping load
// addresses and predicating only the stores.
// ---------------------------------------------------------------------------
__global__ __launch_bounds__(128) void gat_gemm_wmma(
    const float* __restrict__ A,
    const float* __restrict__ W0, const float* __restrict__ b0,
    const float* __restrict__ W1, const float* __restrict__ b1,
    float* __restrict__ h_self, float* __restrict__ h_neigh, int n)
{
  const int wave = threadIdx.x >> 5;
  const int lane = threadIdx.x & 31;
  const int lm = lane & 15;        // column-within-tile / A row-within-tile
  const int lh = lane >> 4;        // selects K pair {0,1} vs {2,3}
  const int r0 = (blockIdx.x * 4 + wave) * 16;

  int arow = r0 + lm; if (arow > n - 1) arow = n - 1;   // clamp, keep EXEC full
  const float* aptr = A + (size_t)arow * DIN + lh * 2;

  const float* wbase[2] = {W0, W1};
  v8f acc[2][8] = {};   // zero-init accumulators: [matrix][column tile]

#pragma unroll 2
  for (int k = 0; k < DIN; k += 4) {
    // A fragment 16x4: lane<16 -> {A[r][k],A[r][k+1]}, lane>=16 -> {A[r][k+2],A[r][k+3]}
    v2f a = *(const v2f*)(aptr + k);
#pragma unroll
    for (int m = 0; m < 2; ++m) {
      // B fragment 4x16 of W^T: B[kk][col] = W[col][kk] -> same float2 pattern
      const float* wp = wbase[m] + (size_t)lm * DIN + k + lh * 2;
#pragma unroll
      for (int c = 0; c < 8; ++c) {
        v2f b = *(const v2f*)(wp + (size_t)(c * 16) * DIN);
        acc[m][c] = __builtin_amdgcn_wmma_f32_16x16x4_f32(
            false, a, false, b, (short)0, acc[m][c], false, false);
      }
    }
  }

  // Epilogue: bias + ReLU + store. C/D layout: VGPR j holds row (r0+j) for
  // lanes 0-15 and row (r0+j+8) for lanes 16-31; column = tile*16 + lm.
  float* outs[2] = {h_self, h_neigh};
  const float* biases[2] = {b0, b1};
#pragma unroll
  for (int m = 0; m < 2; ++m) {
#pragma unroll
    for (int c = 0; c < 8; ++c) {
      const int col = c * 16 + lm;
      const float bias = biases[m][col];
#pragma unroll
      for (int j = 0; j < 8; ++j) {
        int row = r0 + j + lh * 8;
        if (row < n) {
          float v = acc[m][c][j] + bias;
          outs[m][(size_t)row * DOUT + col] = fmaxf(v, 0.f);
        }
      }
    }
  }
}

// ---------------------------------------------------------------------------
// Kernel 2: att_neigh[n,h] = lrelu(<h_neigh[n,h,:], attention[1,h,:]>)
// (att_self is recomputed locally in the aggregation kernel.)
// ---------------------------------------------------------------------------
__global__ void gat_att_neigh(const float* __restrict__ h_neigh,
                              const float* __restrict__ att,
                              float* __restrict__ out, int n)
{
  int idx = blockIdx.x * blockDim.x + threadIdx.x;
  if (idx >= n * NHEAD) return;
  int node = idx >> 2, h = idx & 3;
  const float* x = h_neigh + (size_t)node * DOUT + h * DS;
  const float* w = att + DOUT + h * DS;   // attention[1][h][:]
  float s = 0.f;
#pragma unroll
  for (int j = 0; j < DS; ++j) s += x[j] * w[j];
  out[idx] = lrelu(s);
}

// ---------------------------------------------------------------------------
// Kernel 3: CSR row pointers from sorted edge_row (lower_bound per node).
// ---------------------------------------------------------------------------
__global__ void gat_rowptr(const int* __restrict__ erow,
                           int* __restrict__ rowptr, int n, int e)
{
  int idx = blockIdx.x * blockDim.x + threadIdx.x;
  if (idx > n) return;
  int lo = 0, hi = e;
  while (lo < hi) {
    int mid = (lo + hi) >> 1;
    if (erow[mid] < idx) lo = mid + 1; else hi = mid;
  }
  rowptr[idx] = lo;
}

// ---------------------------------------------------------------------------
// Kernel 4: one wave per node. Lane d = feature index within each head.
//  - att_self logits via wave shuffle reductions
//  - pass 1: lane-parallel segment max over this node's edges
//  - pass 2: softmax weights + weighted aggregation of h_neigh rows
//    (h_neigh is 51 MB -> L2-resident gathers; prefetch next row)
//  - per-head layernorm of agg (b=0) and h_self (b=1), blend * 0.5
// ---------------------------------------------------------------------------
__global__ __launch_bounds__(128) void gat_agg_norm(
    const float* __restrict__ h_self, const float* __restrict__ h_neigh,
    const float* __restrict__ att, const float* __restrict__ att_neigh,
    const int* __restrict__ rowptr, const int* __restrict__ ecol,
    const float* __restrict__ eval,
    const float* __restrict__ scale, const float* __restrict__ offset,
    float* __restrict__ out, int n)
{
  const int wave = threadIdx.x >> 5;
  const int d = threadIdx.x & 31;
  const int node = blockIdx.x * 4 + wave;
  if (node >= n) return;

  float hs[NHEAD], aself[NHEAD];
#pragma unroll
  for (int h = 0; h < NHEAD; ++h) {
    hs[h] = h_self[(size_t)node * DOUT + h * DS + d];
    aself[h] = lrelu(wave_sum(hs[h] * att[h * DS + d]));   // attention[0][h][d]
  }

  const int start = rowptr[node], end = rowptr[node + 1];

  // pass 1: segment max of (aself + att_neigh[col]) per head
  float mx[NHEAD] = {-INFINITY, -INFINITY, -INFINITY, -INFINITY};
  for (int e = start + d; e < end; e += 32) {
    const float* an = att_neigh + (size_t)ecol[e] * NHEAD;
#pragma unroll
    for (int h = 0; h < NHEAD; ++h) mx[h] = fmaxf(mx[h], aself[h] + an[h]);
  }
#pragma unroll
  for (int h = 0; h < NHEAD; ++h) mx[h] = wave_max(mx[h]);

  // pass 2: softmax weights (replicated per lane) + weighted aggregation
  float acc[NHEAD] = {0.f, 0.f, 0.f, 0.f};
  float den[NHEAD] = {0.f, 0.f, 0.f, 0.f};
  for (int e = start; e < end; ++e) {
    int c = ecol[e];
    float ev = eval[e];
    const float* an = att_neigh + (size_t)c * NHEAD;
    const float* hn = h_neigh + (size_t)c * DOUT;
    if (e + 1 < end)
      __builtin_prefetch(h_neigh + (size_t)ecol[e + 1] * DOUT + d * 4, 0, 1);
#pragma unroll
    for (int h = 0; h < NHEAD; ++h) {
      float w = __expf(aself[h] + an[h] - mx[h]) * ev;
      den[h] += w;
      acc[h] += w * hn[h * DS + d];
    }
  }

  // finalize: divide, per-head layernorms (population variance), blend
#pragma unroll
  for (int h = 0; h < NHEAD; ++h) {
    float dn = fmaxf(den[h], 1e-10f);
    float ag = acc[h] / dn;

    float mean0 = wave_sum(ag) * (1.f / 32.f);
    float var0  = wave_sum(ag * ag) * (1.f / 32.f) - mean0 * mean0;
    float y0 = (ag - mean0) * scale[h * DS + d] * rsqrtf(var0 + 1e-9f)
               + offset[h * DS + d];                       // b = 0 (agg)

    float x = hs[h];
    float mean1 = wave_sum(x) * (1.f / 32.f);
    float var1  = wave_sum(x * x) * (1.f / 32.f) - mean1 * mean1;
    float y1 = (x - mean1) * scale[DOUT + h * DS + d] * rsqrtf(var1 + 1e-9f)
               + offset[DOUT + h * DS + d];                // b = 1 (h_self)

    out[(size_t)node * DOUT + h * DS + d] = 0.5f * (y0 + y1);
  }
}

// ---------------------------------------------------------------------------
extern "C" void kernel_launch(void* const* d_in, const int* in_sizes, int n_in,
                              void* d_out, int out_size, void* d_ws, size_t ws_size,
                              hipStream_t stream)
{
  const float* feat  = (const float*)d_in[0];
  const int*   erow  = (const int*)d_in[1];
  const int*   ecol  = (const int*)d_in[2];
  const float* eval  = (const float*)d_in[3];
  const float* W0    = (const float*)d_in[4];
  const float* b0    = (const float*)d_in[5];
  const float* W1    = (const float*)d_in[6];
  const float* b1    = (const float*)d_in[7];
  const float* att   = (const float*)d_in[8];
  const float* scale = (const float*)d_in[9];
  const float* offs  = (const float*)d_in[10];

  const int n = in_sizes[0] / DIN;
  const int e = in_sizes[1];

  // workspace layout: h_self[N*128] | h_neigh[N*128] | att_neigh[N*4] | rowptr[N+1]
  float* h_self    = (float*)d_ws;
  float* h_neigh   = h_self + (size_t)n * DOUT;
  float* attn      = h_neigh + (size_t)n * DOUT;
  int*   rowptr    = (int*)(attn + (size_t)n * NHEAD);

  gat_gemm_wmma<<<(n + 63) / 64, 128, 0, stream>>>(feat, W0, b0, W1, b1,
                                                   h_self, h_neigh, n);
  gat_att_neigh<<<(n * NHEAD + 255) / 256, 256, 0, stream>>>(h_neigh, att, attn, n);
  gat_rowptr<<<(n + 1 + 255) / 256, 256, 0, stream>>>(erow, rowptr, n, e);
  gat_agg_norm<<<(n + 3) / 4, 128, 0, stream>>>(h_self, h_neigh, att, attn,
                                                rowptr, ecol, eval, scale, offs,
                                                (float*)d_out, n);
}